// FurthestPointSampling_3315714753178
// MI455X (gfx1250) — compile-verified
//
#include <hip/hip_runtime.h>
#include <hip/hip_bf16.h>

// Furthest Point Sampling for MI455X (gfx1250, wave32).
//
//  - Prep kernel: repack (B,N,3) xyz into 16B-aligned float4 SoA in d_ws
//    (one global_load_b128 per point in the hot loop instead of 3x b32) and
//    fold the |p|^2 <= 1e-3 filter into temp init (temp = -1e10 filtered,
//    +1e10 otherwise). Since dist >= 0 > -1e10, min(temp, dist) keeps
//    filtered points pinned at -1e10 forever == reference semantics.
//  - Main kernel: one 1024-thread block (32 wave32 waves, fills a WGP) per
//    batch; 512 strictly-sequential iterations; working set (50MB xyz SoA +
//    16MB temp) is fully L2-resident (192MB). Inner scan is manually 4x
//    unrolled with grouped b128/b32 loads so the backend can clause them and
//    overlap L2 latency; global_prefetch_b8 runs once per group for both the
//    point and temp streams. Argmax = wave32 ds-permute butterfly + tiny LDS
//    tree with (max value, min index) lexicographic combine == jnp.argmax
//    first-occurrence semantics. Within a thread the scan order is strictly
//    ascending, so a strict '>' suffices locally (no tie compare).
//  - npoint is a device scalar (graph capture forbids host reads), so B and
//    N are derived on-device; the grid is a fixed block count with a
//    batch-stride loop.

#define FPS_BIG      1.0e10f
#define FPS_NEG_BIG  (-1.0e10f)
#define FPS_MAG_EPS  1.0e-3f

__global__ void fps_prep_kernel(const float* __restrict__ xyz,
                                float4* __restrict__ pts,
                                float* __restrict__ temp,
                                int total, int packed)
{
    int p = blockIdx.x * blockDim.x + threadIdx.x;
    if (p >= total) return;
    float x = xyz[3 * p + 0];
    float y = xyz[3 * p + 1];
    float z = xyz[3 * p + 2];
    float mag = x * x + y * y + z * z;
    if (packed) pts[p] = make_float4(x, y, z, 0.0f);
    temp[p] = (mag <= FPS_MAG_EPS) ? FPS_NEG_BIG : FPS_BIG;
}

template <int BS>
__global__ __launch_bounds__(BS) void fps_main_kernel(
    const float* __restrict__ xyz,     // original interleaved (fallback path)
    const float4* __restrict__ pts,    // packed SoA (preferred path)
    float* __restrict__ temp,
    const int* __restrict__ npoint_ptr,
    int* __restrict__ out,
    int total_pts, int out_total, int packed)
{
    constexpr int NW = BS / 32;        // waves per block (wave32)
    __shared__ float s_val[NW];
    __shared__ int   s_idx[NW];
    __shared__ int   s_far;

    const int npoint = npoint_ptr[0];
    if (npoint <= 0) return;
    const int B = out_total / npoint;
    const int N = total_pts / B;

    const int tid  = threadIdx.x;
    const int lane = tid & 31;
    const int wid  = tid >> 5;

    for (int b = blockIdx.x; b < B; b += gridDim.x) {
        const long long base = (long long)b * (long long)N;
        float*        tb = temp + base;
        const float4* pb = pts + base;
        const float*  xb = xyz + base * 3;
        int*          ob = out + (long long)b * (long long)npoint;

        int far = 0;
        for (int it = 0; it < npoint; ++it) {
            // Emit the index chosen BEFORE this iteration's update
            // (reference emits the scan carry, so out[0] == 0).
            if (tid == 0) ob[it] = far;

            // Broadcast centroid: uniform-address load, hits L0/L2.
            float cx, cy, cz;
            if (packed) {
                float4 c = pb[far];
                cx = c.x; cy = c.y; cz = c.z;
            } else {
                cx = xb[3 * far + 0];
                cy = xb[3 * far + 1];
                cz = xb[3 * far + 2];
            }

            float bv = -__builtin_inff();
            int   bi = 0x7FFFFFFF;

            if (packed) {
                int n = tid;
                // 4x-unrolled main scan: grouped loads -> clause-friendly,
                // overlaps L2 latency; one prefetch pair per group.
                for (; n + 3 * BS < N; n += 4 * BS) {
                    __builtin_prefetch(&pb[n + 16 * BS], 0, 1);  // global_prefetch_b8
                    __builtin_prefetch(&tb[n + 16 * BS], 0, 1);
                    float4 q0 = pb[n + 0 * BS];
                    float4 q1 = pb[n + 1 * BS];
                    float4 q2 = pb[n + 2 * BS];
                    float4 q3 = pb[n + 3 * BS];
                    float  t0 = tb[n + 0 * BS];
                    float  t1 = tb[n + 1 * BS];
                    float  t2 = tb[n + 2 * BS];
                    float  t3 = tb[n + 3 * BS];

                    float dx, dy, dz;
                    dx = q0.x - cx; dy = q0.y - cy; dz = q0.z - cz;
                    t0 = fminf(t0, dx * dx + dy * dy + dz * dz);
                    dx = q1.x - cx; dy = q1.y - cy; dz = q1.z - cz;
                    t1 = fminf(t1, dx * dx + dy * dy + dz * dz);
                    dx = q2.x - cx; dy = q2.y - cy; dz = q2.z - cz;
                    t2 = fminf(t2, dx * dx + dy * dy + dz * dz);
                    dx = q3.x - cx; dy = q3.y - cy; dz = q3.z - cz;
                    t3 = fminf(t3, dx * dx + dy * dy + dz * dz);

                    tb[n + 0 * BS] = t0;
                    tb[n + 1 * BS] = t1;
                    tb[n + 2 * BS] = t2;
                    tb[n + 3 * BS] = t3;

                    // Ascending index order -> strict '>' keeps first max.
                    if (t0 > bv) { bv = t0; bi = n + 0 * BS; }
                    if (t1 > bv) { bv = t1; bi = n + 1 * BS; }
                    if (t2 > bv) { bv = t2; bi = n + 2 * BS; }
                    if (t3 > bv) { bv = t3; bi = n + 3 * BS; }
                }
                for (; n < N; n += BS) {
                    float4 q = pb[n];
                    float dx = q.x - cx, dy = q.y - cy, dz = q.z - cz;
                    float t = fminf(tb[n], dx * dx + dy * dy + dz * dz);
                    tb[n] = t;
                    if (t > bv) { bv = t; bi = n; }
                }
            } else {
                for (int n = tid; n < N; n += BS) {
                    float dx = xb[3 * n + 0] - cx;
                    float dy = xb[3 * n + 1] - cy;
                    float dz = xb[3 * n + 2] - cz;
                    float t = fminf(tb[n], dx * dx + dy * dy + dz * dz);
                    tb[n] = t;
                    if (t > bv) { bv = t; bi = n; }
                }
            }

            // Wave32 butterfly argmax: lexicographic (max value, min index).
            #pragma unroll
            for (int off = 16; off > 0; off >>= 1) {
                float ov = __shfl_xor(bv, off, 32);
                int   oi = __shfl_xor(bi, off, 32);
                if (ov > bv || (ov == bv && oi < bi)) { bv = ov; bi = oi; }
            }
            if (lane == 0) { s_val[wid] = bv; s_idx[wid] = bi; }
            __syncthreads();

            if (wid == 0) {
                float v = (lane < NW) ? s_val[lane] : -__builtin_inff();
                int   i = (lane < NW) ? s_idx[lane] : 0x7FFFFFFF;
                #pragma unroll
                for (int off = 16; off > 0; off >>= 1) {
                    float ov = __shfl_xor(v, off, 32);
                    int   oi = __shfl_xor(i, off, 32);
                    if (ov > v || (ov == v && oi < i)) { v = ov; i = oi; }
                }
                if (lane == 0) s_far = i;
            }
            __syncthreads();
            far = s_far;
        }
        __syncthreads();   // protect s_far/s_val reuse across batch-stride loop
    }
}

extern "C" void kernel_launch(void* const* d_in, const int* in_sizes, int n_in,
                              void* d_out, int out_size, void* d_ws, size_t ws_size,
                              hipStream_t stream)
{
    const float* xyz    = (const float*)d_in[0];
    const int*   npoint = (const int*)d_in[1];   // device scalar
    int*         out    = (int*)d_out;           // int32 indices (4B elems)

    const long long total_pts = (long long)in_sizes[0] / 3;   // B*N

    // ws layout (packed): [float4 pts : total_pts*16B][float temp : total_pts*4B]
    // Fallback (small ws): [float temp : total_pts*4B], read xyz interleaved.
    const int packed = (ws_size >= (size_t)total_pts * 20u) ? 1 : 0;

    float4* pts;
    float*  temp;
    if (packed) {
        pts  = (float4*)d_ws;
        temp = (float*)((char*)d_ws + (size_t)total_pts * sizeof(float4));
    } else {
        pts  = (float4*)d_ws;   // unused in fallback
        temp = (float*)d_ws;
    }

    const int PT = 256;
    const int prep_blocks = (int)((total_pts + PT - 1) / PT);
    fps_prep_kernel<<<prep_blocks, PT, 0, stream>>>(xyz, pts, temp,
                                                    (int)total_pts, packed);

    constexpr int BS = 1024;   // 32 wave32 waves -> fills one WGP per batch
    fps_main_kernel<BS><<<128, BS, 0, stream>>>(xyz, pts, temp, npoint, out,
                                                (int)total_pts, out_size, packed);
}